// SNNBackbone_32701880992465
// MI455X (gfx1250) — compile-verified
//
#include <hip/hip_runtime.h>

typedef __attribute__((ext_vector_type(2))) float v2f;
typedef __attribute__((ext_vector_type(8))) float v8f;

namespace {
constexpr int kL      = 512;   // time steps
constexpr int kN      = 256;   // batch
constexpr int kObs    = 128;   // encoder input features
constexpr int kH      = 512;   // hidden width
constexpr int kLay    = 4;     // FC layers after encoder
constexpr int kStride = kH + 4;  // LDS row stride (floats); 516 % 64 == 4 -> distinct banks per row
constexpr float kBeta = 0.9f;
constexpr float kThr  = 1.0f;
}

__device__ __forceinline__ v8f wmma_f32_k4(v2f a, v2f b, v8f c) {
  // D = A(16x4 f32) * B(4x16 f32) + C(16x16 f32)
  return __builtin_amdgcn_wmma_f32_16x16x4_f32(false, a, false, b, (short)0, c,
                                               false, false);
}

__device__ __forceinline__ v8f splat8(float v) {
  v8f r;
#pragma unroll
  for (int j = 0; j < 8; ++j) r[j] = v;
  return r;
}

// Streaming (read-once) load: keep L2 for the weight working set.
__device__ __forceinline__ v2f ldnt2(const float* p) {
  return __builtin_nontemporal_load((const v2f*)p);
}
// Streaming (write-once) store: don't let the 271 MB output flush L2.
__device__ __forceinline__ void stnt(float* p, float v) {
  __builtin_nontemporal_store(v, p);
}

// One workgroup per 16-row batch slab; 16 waves; each wave owns two 16-col tiles.
// Membrane state persistent in VGPRs across the whole time loop; spikes via LDS.
__global__ __launch_bounds__(512, 1) void snn_fused_kernel(
    const float* __restrict__ x,     // [L, N, OBS]
    const float* __restrict__ mem0,  // [5, N, H]
    const float* __restrict__ encW,  // [H, OBS]
    const float* __restrict__ encB,  // [H]
    const float* __restrict__ fcW,   // [4, H, H]
    const float* __restrict__ fcB,   // [4, H]
    float* __restrict__ out)         // [L*N*H] spikes ++ [5*N*H] mem_final
{
  __shared__ float spk[16 * kStride];  // 33 KB spike staging buffer

  const int tid  = (int)threadIdx.x;
  const int wv   = tid >> 5;   // wave 0..15
  const int lane = tid & 31;
  const int lrow = lane & 15;  // M row (A) / N col (B,C) within tile
  const int lhi  = lane >> 4;  // K half (A,B) / M offset 8 (C)
  const int n0   = (int)blockIdx.x * 16;          // global batch-row base
  const int h0[2] = {wv * 16, (wv + 16) * 16};    // two column tiles per wave

  // Hoisted biases: column-only dependence, constant over time.
  float bias[kLay + 1][2];
#pragma unroll
  for (int tl = 0; tl < 2; ++tl) {
    bias[0][tl] = encB[h0[tl] + lrow];
#pragma unroll
    for (int l = 0; l < kLay; ++l) bias[l + 1][tl] = fcB[l * kH + h0[tl] + lrow];
  }

  // Persistent membrane state, WMMA C/D layout: elem (m = j + 8*lhi, n = lrow).
  v8f mst[kLay + 1][2];
#pragma unroll
  for (int l = 0; l <= kLay; ++l)
#pragma unroll
    for (int tl = 0; tl < 2; ++tl)
#pragma unroll
      for (int j = 0; j < 8; ++j)
        mst[l][tl][j] =
            mem0[((size_t)l * kN + (n0 + j + 8 * lhi)) * kH + h0[tl] + lrow];

  for (int t = 0; t < kL; ++t) {
    // ---------------- layer 0: encoder  x_t @ encW^T + b ----------------
    const float* arow = x + ((size_t)t * kN + (n0 + lrow)) * kObs + 2 * lhi;
    const float* w0p  = encW + (size_t)(h0[0] + lrow) * kObs + 2 * lhi;
    const float* w1p  = encW + (size_t)(h0[1] + lrow) * kObs + 2 * lhi;
    v8f acc0 = splat8(bias[0][0]);
    v8f acc1 = splat8(bias[0][1]);
#pragma unroll 2
    for (int k = 0; k < kObs; k += 4) {
      v2f a  = ldnt2(arow + k);  // read-once activation stream
      v2f b0 = *(const v2f*)(w0p + k);
      v2f b1 = *(const v2f*)(w1p + k);
      acc0 = wmma_f32_k4(a, b0, acc0);
      acc1 = wmma_f32_k4(a, b1, acc1);
    }

    // Prefetch next step's x slab while the FC layers run (hides HBM latency).
    if (t + 1 < kL) {
      const float* nx =
          x + ((size_t)(t + 1) * kN + (n0 + lrow)) * kObs + lhi * 64;
      __builtin_prefetch(nx, 0, 3);
      __builtin_prefetch(nx + 32, 0, 3);
    }

    __syncthreads();  // previous layer's LDS readers are all done
#pragma unroll
    for (int tl = 0; tl < 2; ++tl) {
      v8f acc = tl ? acc1 : acc0;
#pragma unroll
      for (int j = 0; j < 8; ++j) {
        float mp  = mst[0][tl][j];
        float rst = (mp > kThr) ? kThr : 0.0f;   // reset from previous mem
        float mn  = kBeta * mp + acc[j] - rst;   // leaky integrate, subtract reset
        mst[0][tl][j] = mn;
        spk[(j + 8 * lhi) * kStride + h0[tl] + lrow] = (mn > kThr) ? 1.0f : 0.0f;
      }
    }
    __syncthreads();  // spikes visible to all waves

    // ---------------- layers 1..4: spk @ fcW[l]^T + b ----------------
#pragma unroll
    for (int l = 0; l < kLay; ++l) {
      const float* W  = fcW + (size_t)l * kH * kH;
      const float* f0 = W + (size_t)(h0[0] + lrow) * kH + 2 * lhi;
      const float* f1 = W + (size_t)(h0[1] + lrow) * kH + 2 * lhi;
      const int sidx  = lrow * kStride + 2 * lhi;
      v8f a0 = splat8(bias[l + 1][0]);
      v8f a1 = splat8(bias[l + 1][1]);
#pragma unroll 2
      for (int k = 0; k < kH; k += 4) {
        v2f a  = *(const v2f*)&spk[sidx + k];  // ds_load_b64 (A reused x2)
        v2f b0 = *(const v2f*)(f0 + k);        // weights stay L2-resident
        v2f b1 = *(const v2f*)(f1 + k);
        a0 = wmma_f32_k4(a, b0, a0);
        a1 = wmma_f32_k4(a, b1, a1);
      }
      __syncthreads();  // all reads of this layer's input spikes done
      if (l < kLay - 1) {
#pragma unroll
        for (int tl = 0; tl < 2; ++tl) {
          v8f acc = tl ? a1 : a0;
#pragma unroll
          for (int j = 0; j < 8; ++j) {
            float mp  = mst[l + 1][tl][j];
            float rst = (mp > kThr) ? kThr : 0.0f;
            float mn  = kBeta * mp + acc[j] - rst;
            mst[l + 1][tl][j] = mn;
            spk[(j + 8 * lhi) * kStride + h0[tl] + lrow] = (mn > kThr) ? 1.0f : 0.0f;
          }
        }
      } else {  // last layer: spikes go straight to the output tensor
#pragma unroll
        for (int tl = 0; tl < 2; ++tl) {
          v8f acc = tl ? a1 : a0;
#pragma unroll
          for (int j = 0; j < 8; ++j) {
            float mp  = mst[kLay][tl][j];
            float rst = (mp > kThr) ? kThr : 0.0f;
            float mn  = kBeta * mp + acc[j] - rst;
            mst[kLay][tl][j] = mn;
            stnt(&out[((size_t)t * kN + (n0 + j + 8 * lhi)) * kH + h0[tl] + lrow],
                 (mn > kThr) ? 1.0f : 0.0f);
          }
        }
      }
      __syncthreads();
    }
  }

  // Final membrane state dump: out[L*N*H + l*N*H + n*H + h]
  float* mout = out + (size_t)kL * kN * kH;
#pragma unroll
  for (int l = 0; l <= kLay; ++l)
#pragma unroll
    for (int tl = 0; tl < 2; ++tl)
#pragma unroll
      for (int j = 0; j < 8; ++j)
        stnt(&mout[((size_t)l * kN + (n0 + j + 8 * lhi)) * kH + h0[tl] + lrow],
             mst[l][tl][j]);
}

extern "C" void kernel_launch(void* const* d_in, const int* in_sizes, int n_in,
                              void* d_out, int out_size, void* d_ws,
                              size_t ws_size, hipStream_t stream) {
  (void)in_sizes; (void)n_in; (void)out_size; (void)d_ws; (void)ws_size;
  const float* x    = (const float*)d_in[0];
  const float* mem0 = (const float*)d_in[1];
  const float* encW = (const float*)d_in[2];
  const float* encB = (const float*)d_in[3];
  const float* fcW  = (const float*)d_in[4];
  const float* fcB  = (const float*)d_in[5];
  float* out = (float*)d_out;
  // 16 workgroups (one per 16-row batch slab), 512 threads = 16 wave32s each.
  snn_fused_kernel<<<dim3(16), dim3(512), 0, stream>>>(x, mem0, encW, encB,
                                                       fcW, fcB, out);
}